// _UnfusedExperts_81398220194554
// MI455X (gfx1250) — compile-verified
//
#include <hip/hip_runtime.h>
#include <hip/hip_bf16.h>

// ---------------------------------------------------------------------------
// MoE SwiGLU (8 experts, H=2048, I=1408, T=2048, top_k=2) for gfx1250.
// Routed (gather/scatter) formulation; fp32 WMMA (V_WMMA_F32_16X16X4_F32).
// M=32 token tiles: each B fragment in registers feeds two 16x16 WMMA tiles,
// halving weight-fetch traffic per FLOP (kernel is weight-bandwidth bound).
// ---------------------------------------------------------------------------

#define NE      8
#define HIDDEN  2048
#define INTER   1408
#define TOKENS  2048
#define TOPK    2
#define NPAIRS  (TOKENS * TOPK)          // 4096
#define TILE_M  32
#define MAXSLOT 4352                     // >= 4096 + 8*31 rounded up, 32-aligned
#define MAXTILE (MAXSLOT / TILE_M)       // 136

typedef __attribute__((ext_vector_type(2))) float v2f;
typedef __attribute__((ext_vector_type(8))) float v8f;

// ------------------------------ kernel 1: routing ---------------------------
__global__ __launch_bounds__(256) void moe_route_kernel(
    const int* __restrict__ topk_idx,     // [T, K] flat
    const float* __restrict__ topk_w,     // [T, K] flat
    int* __restrict__ meta,               // [0..7]=count, [8..15]=base, [16]=total
    int* __restrict__ tok,                // [MAXSLOT]
    float* __restrict__ cw)               // [MAXSLOT]
{
    __shared__ int cnt[NE];
    __shared__ int off[NE];
    __shared__ int base_s[NE];
    const int tid = threadIdx.x;

    if (tid < NE) { cnt[tid] = 0; off[tid] = 0; }
    __syncthreads();

    for (int p = tid; p < NPAIRS; p += 256)
        atomicAdd(&cnt[topk_idx[p]], 1);

    // initialize all slots as padding
    for (int s = tid; s < MAXSLOT; s += 256) { tok[s] = -1; cw[s] = 0.0f; }
    __syncthreads();

    if (tid == 0) {
        int acc = 0;
        for (int e = 0; e < NE; ++e) {
            base_s[e] = acc;
            meta[e] = cnt[e];
            meta[8 + e] = acc;
            acc += ((cnt[e] + TILE_M - 1) / TILE_M) * TILE_M;  // 32-aligned segment
        }
        meta[16] = acc;                           // total slots (multiple of 32)
    }
    __syncthreads();

    for (int p = tid; p < NPAIRS; p += 256) {
        const int e = topk_idx[p];
        const int pos = atomicAdd(&off[e], 1);
        const int s = base_s[e] + pos;
        tok[s] = p >> 1;                          // token id
        cw[s] = topk_w[p];                        // combine weight
    }
}

// --------------------- kernel 2: gate/up GEMM + SwiGLU ----------------------
// grid.x = MAXTILE (slot tiles of 32), grid.y = INTER/128 (11)
// 256 threads = 8 waves; wave w owns columns [n0, n0+16); 2 M-subtiles/wave.
#define K2_CHUNK 256
#define K2_LROW  (K2_CHUNK + 4)          // stride 260 (== 4 mod 64 banks): conflict-free

__global__ __launch_bounds__(256) void moe_gate_up_kernel(
    const float* __restrict__ X,          // [T, H]
    const float* __restrict__ gate_w,     // [E, H, I]
    const float* __restrict__ up_w,       // [E, H, I]
    const int* __restrict__ meta,
    const int* __restrict__ tok,
    float* __restrict__ act)              // [MAXSLOT, I]
{
    __shared__ float lx[TILE_M * K2_LROW];   // 33,280 B
    __shared__ int stok[TILE_M];

    const int slot0 = blockIdx.x * TILE_M;
    const int total = meta[16];
    if (slot0 >= total) return;

    // find owning expert: largest e with base[e] <= slot0
    int e = 0;
#pragma unroll
    for (int i = 1; i < NE; ++i)
        if (slot0 >= meta[8 + i]) e = i;

    const int tid = threadIdx.x;
    if (tid < TILE_M) stok[tid] = tok[slot0 + tid];

    const int wave = tid >> 5;
    const int lane = tid & 31;
    const int ln   = lane & 15;           // N within tile / M for A
    const int kh   = lane >> 4;           // upper lane half: K+2 in A/B frags
    const int n0   = blockIdx.y * 128 + wave * 16;

    const size_t wbase = (size_t)e * HIDDEN * INTER + (size_t)n0 + ln;
    const float* gB = gate_w + wbase;
    const float* uB = up_w + wbase;

    v8f cg0 = {}, cg1 = {};
    v8f cu0 = {}, cu1 = {};

    for (int kc = 0; kc < HIDDEN; kc += K2_CHUNK) {
        __syncthreads();
        // stage X tile chunk [32 x 256] into LDS (coalesced float4 loads)
        for (int q = tid; q < TILE_M * (K2_CHUNK / 4); q += 256) {
            const int r = q / (K2_CHUNK / 4);
            const int c = q % (K2_CHUNK / 4);
            const int t = stok[r];
            float4 v = make_float4(0.f, 0.f, 0.f, 0.f);
            if (t >= 0)
                v = *(const float4*)(X + (size_t)t * HIDDEN + kc + c * 4);
            *(float4*)(&lx[r * K2_LROW + c * 4]) = v;
        }
        __syncthreads();

        const float* lrow0 = &lx[ln * K2_LROW + 2 * kh];
        const float* lrow1 = lrow0 + 16 * K2_LROW;
#pragma unroll 4
        for (int k = 0; k < K2_CHUNK; k += 4) {
            v2f a0 = *(const v2f*)(lrow0 + k);          // rows 0..15
            v2f a1 = *(const v2f*)(lrow1 + k);          // rows 16..31
            const float* gp = gB + (size_t)(kc + k + 2 * kh) * INTER;
            const float* up = uB + (size_t)(kc + k + 2 * kh) * INTER;
            v2f bg; bg.x = gp[0]; bg.y = gp[INTER];
            v2f bu; bu.x = up[0]; bu.y = up[INTER];
            cg0 = __builtin_amdgcn_wmma_f32_16x16x4_f32(
                false, a0, false, bg, (short)0, cg0, false, false);
            cg1 = __builtin_amdgcn_wmma_f32_16x16x4_f32(
                false, a1, false, bg, (short)0, cg1, false, false);
            cu0 = __builtin_amdgcn_wmma_f32_16x16x4_f32(
                false, a0, false, bu, (short)0, cu0, false, false);
            cu1 = __builtin_amdgcn_wmma_f32_16x16x4_f32(
                false, a1, false, bu, (short)0, cu1, false, false);
        }
    }

    // epilogue: act = silu(g) * u  (two M-subtiles)
#pragma unroll
    for (int j = 0; j < 8; ++j) {
        const int m = j + 8 * kh;
        {
            const float g = cg0[j], u = cu0[j];
            const float s = g / (1.0f + __expf(-g));
            act[(size_t)(slot0 + m) * INTER + n0 + ln] = s * u;
        }
        {
            const float g = cg1[j], u = cu1[j];
            const float s = g / (1.0f + __expf(-g));
            act[(size_t)(slot0 + 16 + m) * INTER + n0 + ln] = s * u;
        }
    }
}

// ----------------------- kernel 3: down GEMM + scatter ----------------------
// grid.x = MAXTILE, grid.y = HIDDEN/128 (16)
#define K3_CHUNK 352
#define K3_LROW  (K3_CHUNK + 4)          // stride 356 (== 36 mod 64): conflict-free

__global__ __launch_bounds__(256) void moe_down_kernel(
    const float* __restrict__ act,        // [MAXSLOT, I]
    const float* __restrict__ down_w,     // [E, I, H]
    const int* __restrict__ meta,
    const int* __restrict__ tok,
    const float* __restrict__ cw,
    float* __restrict__ out)              // [T, H] (pre-zeroed)
{
    __shared__ float la[TILE_M * K3_LROW];   // 45,568 B
    __shared__ int stok[TILE_M];
    __shared__ float scw[TILE_M];

    const int slot0 = blockIdx.x * TILE_M;
    const int total = meta[16];
    if (slot0 >= total) return;

    int e = 0;
#pragma unroll
    for (int i = 1; i < NE; ++i)
        if (slot0 >= meta[8 + i]) e = i;

    const int tid = threadIdx.x;
    if (tid < TILE_M) {
        stok[tid] = tok[slot0 + tid];
        scw[tid]  = cw[slot0 + tid];
    }

    const int wave = tid >> 5;
    const int lane = tid & 31;
    const int ln   = lane & 15;
    const int kh   = lane >> 4;
    const int n0   = blockIdx.y * 128 + wave * 16;   // H column

    const float* dB = down_w + (size_t)e * INTER * HIDDEN + (size_t)n0 + ln;

    v8f c0 = {}, c1 = {};

    for (int kc = 0; kc < INTER; kc += K3_CHUNK) {
        __syncthreads();
        // stage act tile chunk [32 x 352] (coalesced float4 loads)
        for (int q = tid; q < TILE_M * (K3_CHUNK / 4); q += 256) {
            const int r = q / (K3_CHUNK / 4);
            const int cc = q % (K3_CHUNK / 4);
            float4 v = *(const float4*)(act + (size_t)(slot0 + r) * INTER + kc + cc * 4);
            *(float4*)(&la[r * K3_LROW + cc * 4]) = v;
        }
        __syncthreads();

        const float* lrow0 = &la[ln * K3_LROW + 2 * kh];
        const float* lrow1 = lrow0 + 16 * K3_LROW;
#pragma unroll 4
        for (int k = 0; k < K3_CHUNK; k += 4) {
            v2f a0 = *(const v2f*)(lrow0 + k);
            v2f a1 = *(const v2f*)(lrow1 + k);
            const float* dp = dB + (size_t)(kc + k + 2 * kh) * HIDDEN;
            v2f b; b.x = dp[0]; b.y = dp[HIDDEN];
            c0 = __builtin_amdgcn_wmma_f32_16x16x4_f32(
                false, a0, false, b, (short)0, c0, false, false);
            c1 = __builtin_amdgcn_wmma_f32_16x16x4_f32(
                false, a1, false, b, (short)0, c1, false, false);
        }
    }

    // epilogue: out[t, h] += cw * y   (exactly 2 adds per element -> deterministic)
#pragma unroll
    for (int j = 0; j < 8; ++j) {
        const int m = j + 8 * kh;
        {
            const int t = stok[m];
            if (t >= 0)
                atomicAdd(&out[(size_t)t * HIDDEN + n0 + ln], scw[m] * c0[j]);
        }
        {
            const int t = stok[16 + m];
            if (t >= 0)
                atomicAdd(&out[(size_t)t * HIDDEN + n0 + ln], scw[16 + m] * c1[j]);
        }
    }
}

// ------------------------------- launcher -----------------------------------
extern "C" void kernel_launch(void* const* d_in, const int* in_sizes, int n_in,
                              void* d_out, int out_size, void* d_ws, size_t ws_size,
                              hipStream_t stream) {
    const float* hidden  = (const float*)d_in[0];   // [T, H]
    const int*   tk_idx  = (const int*)d_in[1];     // [T, K]
    const float* tk_w    = (const float*)d_in[2];   // [T, K]
    const float* gate_w  = (const float*)d_in[3];   // [E, H, I]
    const float* up_w    = (const float*)d_in[4];   // [E, H, I]
    const float* down_w  = (const float*)d_in[5];   // [E, I, H]
    float* out = (float*)d_out;

    // workspace layout
    char* ws = (char*)d_ws;
    int*   meta = (int*)ws;                              // 32 ints (128 B reserved)
    int*   tok  = (int*)(ws + 128);                      // MAXSLOT ints
    float* cwp  = (float*)(ws + 128 + MAXSLOT * 4);      // MAXSLOT floats
    float* act  = (float*)(ws + 128 + MAXSLOT * 8);      // MAXSLOT * INTER floats

    // zero output (atomic-accumulated)
    hipMemsetAsync(d_out, 0, (size_t)out_size * sizeof(float), stream);

    moe_route_kernel<<<1, 256, 0, stream>>>(tk_idx, tk_w, meta, tok, cwp);

    moe_gate_up_kernel<<<dim3(MAXTILE, INTER / 128), 256, 0, stream>>>(
        hidden, gate_w, up_w, meta, tok, act);

    moe_down_kernel<<<dim3(MAXTILE, HIDDEN / 128), 256, 0, stream>>>(
        act, down_w, meta, tok, cwp, out);
}